// GCN_body_86998857548332
// MI455X (gfx1250) — compile-verified
//
#include <hip/hip_runtime.h>

typedef __attribute__((ext_vector_type(2))) float v2f;
typedef __attribute__((ext_vector_type(8))) float v8f;

#define N_NODES 100000
#define N_EDGES 1600000
#define NFEAT   128     // K for both layers (layer2 input = 2*NHID = 128)
#define H1      128
#define H2      64

static __device__ __forceinline__ v8f wmma_f32(v2f a, v2f b, v8f c) {
    return __builtin_amdgcn_wmma_f32_16x16x4_f32(false, a, false, b,
                                                 (short)0, c, false, false);
}

// ---------------------------------------------------------------------------
// Zero-fill, float4-vectorized (n is always a multiple of 4: N*128, N*64)
// ---------------------------------------------------------------------------
__global__ void fill_zero_kernel(float4* __restrict__ p, int n4) {
    int i = blockIdx.x * blockDim.x + threadIdx.x;
    if (i < n4) p[i] = make_float4(0.f, 0.f, 0.f, 0.f);
}

// ---------------------------------------------------------------------------
// C[N x NOUT] = A[N x 128] @ B[128 x NOUT], exact fp32 via V_WMMA_F32_16X16X4_F32.
//
// Tiling: each wave owns a 32 x (16*NBLK) macro-tile (2 row tiles, NBLK col
// tiles). Every loaded A pair and B pair feeds two WMMAs (4 loads : 4 WMMAs
// for NBLK=2). NOUT is a template constant so all B strides fold to immediate
// offsets; all index math is int32. 100000/32 = 3125 blocks exactly and all
// branches are wave-uniform, so EXEC is all-1s for every WMMA.
//
// Layer 1 (NOUT=128): 4 waves x 2 col tiles = 8 tiles -> grid.y folded away.
// Layer 2 (NOUT=64):  4 waves x 1 col tile  = 4 tiles.
//
// A layout (16x4 f32, ISA 7.12.2): lanes 0-15 -> M=lane, K={0,1};
// lanes 16-31 -> M=lane-16, K={2,3}. B mirrors with N across lanes.
// D layout: VGPR r, lanes 0-15 -> D[M=r][N=lane]; lanes 16-31 -> D[M=8+r][N=lane-16].
// ---------------------------------------------------------------------------
template <int NOUT>
__global__ void gemm_f32_wmma_kernel(const float* __restrict__ A,
                                     const float* __restrict__ B,
                                     float* __restrict__ C) {
    constexpr int NBLK = (NOUT == 128) ? 2 : 1;             // col tiles per wave

    const int lane    = threadIdx.x & 31;
    const int wave    = threadIdx.x >> 5;
    const int rowBase = blockIdx.x << 5;                    // 32 rows per wave
    const int colBase = (wave * NBLK) << 4;
    const int m       = lane & 15;
    const int kq      = (lane >> 4) << 1;                   // 0 (lanes 0-15) or 2

    const float* __restrict__ arow0 = A + (rowBase + m) * NFEAT;   // row tile 0
    const float* __restrict__ arow1 = arow0 + 16 * NFEAT;          // row tile 1
    const float* __restrict__ bcol  = B + colBase + m;

    v8f c00 = {}, c10 = {};   // col tile 0, row tiles 0/1
    v8f c01 = {}, c11 = {};   // col tile 1 (NBLK==2 only)

#pragma unroll
    for (int k = 0; k < NFEAT; k += 4) {
        v2f a0, a1, b0;
        a0.x = arow0[k + kq];
        a0.y = arow0[k + kq + 1];
        a1.x = arow1[k + kq];
        a1.y = arow1[k + kq + 1];
        b0.x = bcol[(k + kq) * NOUT];          // compile-time stride -> imm offset
        b0.y = bcol[(k + kq + 1) * NOUT];
        c00 = wmma_f32(a0, b0, c00);
        c10 = wmma_f32(a1, b0, c10);
        if constexpr (NBLK == 2) {
            v2f b1;
            b1.x = bcol[(k + kq) * NOUT + 16];
            b1.y = bcol[(k + kq + 1) * NOUT + 16];
            c01 = wmma_f32(a0, b1, c01);
            c11 = wmma_f32(a1, b1, c11);
        }
    }

    const int mOut = (lane >> 4) << 3;                      // 0 or 8
    float* __restrict__ crow0 = C + (rowBase + mOut) * NOUT + colBase + m;
    float* __restrict__ crow1 = crow0 + 16 * NOUT;
#pragma unroll
    for (int r = 0; r < 8; ++r) {
        crow0[r * NOUT] = c00[r];
        crow1[r * NOUT] = c10[r];
        if constexpr (NBLK == 2) {
            crow0[r * NOUT + 16] = c01[r];
            crow1[r * NOUT + 16] = c11[r];
        }
    }
}

// ---------------------------------------------------------------------------
// Edge-parallel SpMM scatter: one wave32 per edge.
// The edge id is forced wave-uniform with readfirstlane so src/dst/weight
// compile to scalar (SMEM) loads and the row bases become SGPR bases (GVS
// addressing: SGPR base + lane offset) instead of 32 redundant vector loads.
// F=128: each lane gathers one float4 (global_load_b128, 512B/wave coalesced)
// and issues 4 global_atomic_add_f32; F=64: float2 + 2 atomics. Node arrays
// are L2-resident (<= 51 MB vs 192 MB L2) -> pure L2 atomic throughput.
// ---------------------------------------------------------------------------
template <int F>
__global__ void spmm_scatter_kernel(const int* __restrict__ src,
                                    const int* __restrict__ dst,
                                    const float* __restrict__ ew,
                                    const float* __restrict__ support,
                                    float* __restrict__ agg) {
    const int lane = threadIdx.x & 31;
    const int e = __builtin_amdgcn_readfirstlane(
        (blockIdx.x * blockDim.x + threadIdx.x) >> 5);      // wave-uniform edge id
    if (e >= N_EDGES) return;                               // uniform branch

    const int   s = src[e];
    const int   d = dst[e];
    const float w = ew[e];

    constexpr int VEC = F / 32;                 // 4 (F=128) or 2 (F=64)
    const float* __restrict__ srow = support + s * F + lane * VEC;
    float* __restrict__       drow = agg + d * F + lane * VEC;

    if constexpr (VEC == 4) {
        const float4 v = *(const float4*)srow;
        atomicAdd(drow + 0, w * v.x);
        atomicAdd(drow + 1, w * v.y);
        atomicAdd(drow + 2, w * v.z);
        atomicAdd(drow + 3, w * v.w);
    } else {
        const float2 v = *(const float2*)srow;
        atomicAdd(drow + 0, w * v.x);
        atomicAdd(drow + 1, w * v.y);
    }
}

// ---------------------------------------------------------------------------
// out = relu(agg + bias), float4-vectorized. Row has F/4 float4 chunks;
// fmask4 = F/4 - 1 selects the bias chunk.
// ---------------------------------------------------------------------------
__global__ void bias_relu_kernel(const float4* __restrict__ agg,
                                 const float4* __restrict__ bias,
                                 float4* __restrict__ out, int fmask4, int n4) {
    int i = blockIdx.x * blockDim.x + threadIdx.x;
    if (i >= n4) return;
    const float4 a = agg[i];
    const float4 b = bias[i & fmask4];
    out[i] = make_float4(fmaxf(a.x + b.x, 0.f), fmaxf(a.y + b.y, 0.f),
                         fmaxf(a.z + b.z, 0.f), fmaxf(a.w + b.w, 0.f));
}

extern "C" void kernel_launch(void* const* d_in, const int* in_sizes, int n_in,
                              void* d_out, int out_size, void* d_ws, size_t ws_size,
                              hipStream_t stream) {
    (void)in_sizes; (void)n_in; (void)out_size; (void)ws_size;

    const float* x   = (const float*)d_in[0];   // [N_NODES, 128]
    const int*   ei  = (const int*)d_in[1];     // [2, N_EDGES]
    const float* ew  = (const float*)d_in[2];   // [N_EDGES]
    const float* W1  = (const float*)d_in[3];   // [128, 128]
    const float* b1  = (const float*)d_in[4];   // [128]
    const float* W2  = (const float*)d_in[5];   // [128, 64]
    const float* b2  = (const float*)d_in[6];   // [64]
    float*       out = (float*)d_out;           // [N_NODES, 64]

    const int* srcI = ei;
    const int* dstI = ei + N_EDGES;

    // Workspace layout: sup (N*128 f32) | agg (N*128 f32)  -> 102.4 MB total.
    float* sup = (float*)d_ws;
    float* agg = sup + (size_t)N_NODES * 128;

    const int n1_4 = N_NODES * H1 / 4;               // 3.2M float4
    const int n2_4 = N_NODES * H2 / 4;               // 1.6M float4
    const dim3 gemmBlk(128);                         // 4 waves
    const int  gemmGridX     = N_NODES / 32;         // 3125 exact
    const int  scatterBlocks = (N_EDGES * 32) / 256; // one wave per edge, exact

    // ---------------- Layer 1: 128 -> 128 ----------------
    fill_zero_kernel<<<(n1_4 + 255) / 256, 256, 0, stream>>>((float4*)agg, n1_4);
    gemm_f32_wmma_kernel<H1><<<gemmGridX, gemmBlk, 0, stream>>>(x, W1, sup);
    spmm_scatter_kernel<H1><<<scatterBlocks, 256, 0, stream>>>(srcI, dstI, ew, sup, agg);
    bias_relu_kernel<<<(n1_4 + 255) / 256, 256, 0, stream>>>(
        (const float4*)agg, (const float4*)b1, (float4*)agg, H1 / 4 - 1, n1_4); // h1 in place

    // ---------------- Layer 2: 128 -> 64 -----------------
    // Stream order serializes kernels: the GEMM reads h1 (agg) fully before
    // the fill reuses that region as the layer-2 accumulator.
    gemm_f32_wmma_kernel<H2><<<gemmGridX, gemmBlk, 0, stream>>>(agg, W2, sup);
    fill_zero_kernel<<<(n2_4 + 255) / 256, 256, 0, stream>>>((float4*)agg, n2_4);
    spmm_scatter_kernel<H2><<<scatterBlocks, 256, 0, stream>>>(srcI, dstI, ew, sup, agg);
    bias_relu_kernel<<<(n2_4 + 255) / 256, 256, 0, stream>>>(
        (const float4*)agg, (const float4*)b2, (float4*)out, H2 / 4 - 1, n2_4);
}